// MACELayer_49520972923318
// MI455X (gfx1250) — compile-verified
//
#include <hip/hip_runtime.h>
#include <math.h>

// ---------------------------------------------------------------------------
// MACE layer (0e+1o, F=64) on gfx1250: fp32 WMMA 16x16x4 for all dense GEMMs,
// fused edge kernel (radial MLP -> CG tensor product -> atomic scatter).
// Round 2: register-blocked B operand in edge GEMM (2x less W_rad2 L2 traffic),
// vectorized (b64) A-fragment loads for unit-stride operands.
// ---------------------------------------------------------------------------

typedef __attribute__((ext_vector_type(2))) float v2f;
typedef __attribute__((ext_vector_type(8))) float v8f;

#define NN 20000
#define EE 320000
#define FF 64
#define NF (NN * FF)
#define PF 320   // P*F
#define EPB 32   // edges per block in k_edge

// One 16x16 output tile of C = A[16x64] * B[64xN], via 16 chained
// V_WMMA_F32_16X16X4_F32. A element (m,k) at A[m*lda + k*as]; B element (k,n)
// at B[k*ldb + n]. AVEC: A K-stride is 1 and rows 8B-aligned -> b64 loads.
// Wave32, EXEC all-ones at every call site (no divergence).
template <bool AVEC>
__device__ __forceinline__ v8f wmma_tile_k64(const float* A, int lda, int as,
                                             const float* B, int ldb, int lane)
{
    v8f acc = {};
    const int l15  = lane & 15;
    const int koff = (lane >> 4) << 1;   // lanes 0-15 -> K+0/K+1, lanes 16-31 -> K+2/K+3
#pragma unroll
    for (int k0 = 0; k0 < 64; k0 += 4) {
        const int ka = k0 + koff;
        v2f a, b;
        if (AVEC) {
            a = *(const v2f*)(A + l15 * lda + ka);          // contiguous pair, 8B aligned
        } else {
            a.x = A[l15 * lda + ka * as];
            a.y = A[l15 * lda + (ka + 1) * as];
        }
        b.x = B[ka * ldb + l15];
        b.y = B[(ka + 1) * ldb + l15];
        acc = __builtin_amdgcn_wmma_f32_16x16x4_f32(false, a, false, b,
                                                    (short)0, acc, false, false);
    }
    return acc;
}

// ---------------------------------------------------------------------------
// K1: s1 = s @ W_lin_s ; v1[c] = v[:,:,c] @ W_lin_v  (c = 0..2, planar out)
// grid = N/16 node tiles, 512 threads = 16 waves, 1 wave per (mat, col-tile).
// ---------------------------------------------------------------------------
__global__ void __launch_bounds__(512)
k_node_linear(const float* __restrict__ s, const float* __restrict__ v,
              const float* __restrict__ Wls, const float* __restrict__ Wlv,
              float* __restrict__ s1, float* __restrict__ v1)
{
    const int tid  = threadIdx.x;
    const int lane = tid & 31;
    const int w    = tid >> 5;      // 0..15
    const int mat  = w >> 2;        // 0 = scalar, 1..3 = vector plane
    const int nt   = w & 3;         // 16-col tile
    const int m0   = blockIdx.x * 16;

    v8f acc;
    float* O;
    if (mat == 0) {
        acc = wmma_tile_k64<true>(s + (size_t)m0 * FF, FF, 1, Wls + nt * 16, FF, lane);
        O = s1;
    } else {
        int c = mat - 1;
        acc = wmma_tile_k64<false>(v + (size_t)m0 * FF * 3 + c, FF * 3, 3,
                                   Wlv + nt * 16, FF, lane);
        O = v1 + (size_t)c * NF;
    }

    float* C = O + (size_t)m0 * FF + nt * 16;
    const int l15 = lane & 15;
    const int rb  = (lane < 16) ? 0 : 8;
#pragma unroll
    for (int r = 0; r < 8; ++r) C[(rb + r) * FF + l15] = acc[r];
}

// ---------------------------------------------------------------------------
// K1b: species-dependent skip: sc_s[n,g] = sum_f s[n,f] W_sc_s[sp,f,g] (+v x3)
// ---------------------------------------------------------------------------
__global__ void __launch_bounds__(256)
k_skip(const float* __restrict__ s, const float* __restrict__ v,
       const int* __restrict__ spec,
       const float* __restrict__ Wscs, const float* __restrict__ Wscv,
       float* __restrict__ sc_s, float* __restrict__ sc_v)
{
    int idx = blockIdx.x * 256 + threadIdx.x;
    if (idx >= NF) return;
    int n = idx >> 6, g = idx & 63;
    const float* Ws = Wscs + (size_t)spec[n] * 4096;
    const float* Wv = Wscv + (size_t)spec[n] * 4096;
    const float* sn = s + (size_t)n * 64;
    const float* vn = v + (size_t)n * 192;
    float a0 = 0.f, a1 = 0.f, a2 = 0.f, a3 = 0.f;
    for (int f = 0; f < 64; ++f) {
        float ws = Ws[f * 64 + g], wv = Wv[f * 64 + g];
        a0 += sn[f] * ws;
        a1 += vn[f * 3 + 0] * wv;
        a2 += vn[f * 3 + 1] * wv;
        a3 += vn[f * 3 + 2] * wv;
    }
    sc_s[idx] = a0;
    sc_v[idx] = a1; sc_v[NF + idx] = a2; sc_v[2 * NF + idx] = a3;
}

// ---------------------------------------------------------------------------
// K2: fused edge kernel. Block = 128 threads (4 waves), 32 edges per block.
//   stage 1: h = silu(ef @ W_rad1)            -> LDS            (VALU)
//   stage 2: tw[32,320] = h @ W_rad2           -> LDS via WMMA.
//            Each wave owns 5 column tiles; B fragments (16 x v2f = 32 VGPRs)
//            are loaded once per column tile and reused across 2 M-tiles.
//   stage 3: CG tensor product + atomic scatter to receivers.
// ---------------------------------------------------------------------------
__global__ void __launch_bounds__(128)
k_edge(const float* __restrict__ Y1, const float* __restrict__ ef,
       const int* __restrict__ snd, const int* __restrict__ rcv,
       const float* __restrict__ W1, const float* __restrict__ W2,
       const float* __restrict__ s1, const float* __restrict__ v1,
       float* __restrict__ As, float* __restrict__ Av)
{
    __shared__ float sh_ef[EPB][8];     // 1 KB
    __shared__ float sh_h[EPB][64];     // 8 KB
    __shared__ float sh_tw[EPB][PF];    // 40 KB
    __shared__ float sh_y[EPB][3];
    __shared__ int   sh_snd[EPB], sh_rcv[EPB];

    const int tid  = threadIdx.x;
    const int base = blockIdx.x * EPB;

    for (int i = tid; i < EPB * 8; i += 128)
        sh_ef[i >> 3][i & 7] = ef[(size_t)base * 8 + i];
    for (int i = tid; i < EPB * 3; i += 128)
        sh_y[i / 3][i % 3] = Y1[(size_t)base * 3 + i];
    if (tid < EPB) { sh_snd[tid] = snd[base + tid]; sh_rcv[tid] = rcv[base + tid]; }
    __syncthreads();

    // ---- stage 1: radial layer 1 + SiLU (2048 outputs / 128 threads) ----
#pragma unroll
    for (int i = 0; i < 16; ++i) {
        int o = tid + i * 128;
        int e = o >> 6, j = o & 63;
        float x = 0.f;
#pragma unroll
        for (int r = 0; r < 8; ++r) x += sh_ef[e][r] * W1[r * 64 + j];
        sh_h[e][j] = x / (1.f + __expf(-x));
    }
    __syncthreads();

    // ---- stage 2: tw = h @ W_rad2 via WMMA, B register-blocked ----
    {
        const int lane = tid & 31;
        const int w    = tid >> 5;           // 0..3
        const int l15  = lane & 15;
        const int koff = (lane >> 4) << 1;
        const int rb   = (lane < 16) ? 0 : 8;
        for (int t = 0; t < 5; ++t) {
            const int nt = w * 5 + t;
            const float* Bp = W2 + nt * 16;
            v2f bfr[16];
#pragma unroll
            for (int kk = 0; kk < 16; ++kk) {    // load all B fragments once
                int ka = kk * 4 + koff;
                bfr[kk].x = Bp[ka * PF + l15];
                bfr[kk].y = Bp[(ka + 1) * PF + l15];
            }
#pragma unroll
            for (int mt = 0; mt < EPB / 16; ++mt) {   // reuse B across M-tiles
                v8f acc = {};
#pragma unroll
                for (int kk = 0; kk < 16; ++kk) {
                    int ka = kk * 4 + koff;
                    v2f a = *(const v2f*)&sh_h[mt * 16 + l15][ka];  // ds b64
                    acc = __builtin_amdgcn_wmma_f32_16x16x4_f32(
                              false, a, false, bfr[kk], (short)0, acc, false, false);
                }
                float* C = &sh_tw[mt * 16][nt * 16];
#pragma unroll
                for (int r = 0; r < 8; ++r) C[(rb + r) * PF + l15] = acc[r];
            }
        }
    }
    __syncthreads();

    // ---- stage 3: messages + atomic scatter (2 edges x 64 channels / iter) ----
    const float IS3 = 0.57735026919f;   // 1/sqrt(3)
    const float IS2 = 0.70710678119f;   // 1/sqrt(2)
    const int f  = tid & 63;
    const int eh = tid >> 6;
    for (int eo = 0; eo < EPB; eo += 2) {
        int e  = eo + eh;
        int gs = sh_snd[e], gr = sh_rcv[e];
        if (eo + 2 < EPB) {  // software-pipeline: prefetch next gather rows
            int gs2 = sh_snd[eo + 2 + eh];
            __builtin_prefetch(&s1[(size_t)gs2 * 64 + f], 0, 3);
            __builtin_prefetch(&v1[(size_t)gs2 * 64 + f], 0, 3);
            __builtin_prefetch(&v1[NF + (size_t)gs2 * 64 + f], 0, 3);
            __builtin_prefetch(&v1[2 * NF + (size_t)gs2 * 64 + f], 0, 3);
        }
        float ss = s1[(size_t)gs * 64 + f];
        float vx = v1[(size_t)gs * 64 + f];
        float vy = v1[NF + (size_t)gs * 64 + f];
        float vz = v1[2 * NF + (size_t)gs * 64 + f];
        float yx = sh_y[e][0], yy = sh_y[e][1], yz = sh_y[e][2];
        float dot = vx * yx + vy * yy + vz * yz;
        float cx = vy * yz - vz * yy;
        float cy = vz * yx - vx * yz;
        float cz = vx * yy - vy * yx;
        float t0 = sh_tw[e][f],        t1 = sh_tw[e][64 + f],
              t2 = sh_tw[e][128 + f],  t3 = sh_tw[e][192 + f],
              t4 = sh_tw[e][256 + f] * IS2;
        float ms = t0 * ss + t1 * dot * IS3;
        float mx = t2 * ss * yx + t3 * vx + t4 * cx;
        float my = t2 * ss * yy + t3 * vy + t4 * cy;
        float mz = t2 * ss * yz + t3 * vz + t4 * cz;
        atomicAdd(&As[(size_t)gr * 64 + f], ms);
        atomicAdd(&Av[(size_t)gr * 64 + f], mx);
        atomicAdd(&Av[NF + (size_t)gr * 64 + f], my);
        atomicAdd(&Av[2 * NF + (size_t)gr * 64 + f], mz);
    }
}

// ---------------------------------------------------------------------------
// K3: post-aggregation linear: Asp = (A_s @ W_int_s)*inv, Avp planes likewise.
// Writes into the (now dead) s1 / v1 buffers.
// ---------------------------------------------------------------------------
__global__ void __launch_bounds__(512)
k_int_linear(const float* __restrict__ As, const float* __restrict__ Av,
             const float* __restrict__ Wis, const float* __restrict__ Wiv,
             float* __restrict__ Os, float* __restrict__ Ov)
{
    const float inv = 0.25f;  // 1/sqrt(AVG_NEIGH=16)
    const int tid  = threadIdx.x;
    const int lane = tid & 31;
    const int w    = tid >> 5;
    const int mat  = w >> 2;
    const int nt   = w & 3;
    const int m0   = blockIdx.x * 16;

    const float* A; const float* B; float* O;
    if (mat == 0) { A = As + (size_t)m0 * FF; B = Wis; O = Os; }
    else { int c = mat - 1;
           A = Av + (size_t)c * NF + (size_t)m0 * FF; B = Wiv;
           O = Ov + (size_t)c * NF; }

    v8f acc = wmma_tile_k64<true>(A, FF, 1, B + nt * 16, FF, lane);

    float* C = O + (size_t)m0 * FF + nt * 16;
    const int l15 = lane & 15;
    const int rb  = (lane < 16) ? 0 : 8;
#pragma unroll
    for (int r = 0; r < 8; ++r) C[(rb + r) * FF + l15] = acc[r] * inv;
}

// ---------------------------------------------------------------------------
// K4a: symmetric-contraction product basis (elementwise, species weights).
// Writes B_s / B_v into the (now dead) A_s / A_v buffers.
// ---------------------------------------------------------------------------
__global__ void __launch_bounds__(256)
k_product(const float* __restrict__ Asp, const float* __restrict__ Avp,
          const int* __restrict__ spec,
          const float* __restrict__ wps, const float* __restrict__ wpv,
          float* __restrict__ Bs, float* __restrict__ Bv)
{
    int idx = blockIdx.x * 256 + threadIdx.x;
    if (idx >= NF) return;
    int n = idx >> 6, f = idx & 63;
    float a = Asp[idx];
    float x = Avp[idx], y = Avp[NF + idx], z = Avp[2 * NF + idx];
    float d = x * x + y * y + z * z;
    const float* ws = wps + (size_t)spec[n] * (5 * 64);
    const float* wv = wpv + (size_t)spec[n] * (4 * 64);
    float a2 = a * a;
    float bs = ws[f] * a + ws[64 + f] * a2 + ws[128 + f] * d
             + ws[192 + f] * a2 * a + ws[256 + f] * a * d;
    float gv = wv[f] + wv[64 + f] * a + wv[128 + f] * a2 + wv[192 + f] * d;
    Bs[idx] = bs;
    Bv[idx] = gv * x; Bv[NF + idx] = gv * y; Bv[2 * NF + idx] = gv * z;
}

// ---------------------------------------------------------------------------
// K4b: s_out = B_s @ W_prod_s + sc_s ; v_out = B_v @ W_prod_v + sc_v.
// s_out -> d_out[N .. N+NF) and (in-place-safe) into the sc_s buffer for the
// readout; v_out -> d_out interleaved [N,F,3].  NOTE: souts aliases sc_s by
// design (each element is read then written by the same thread) -> no restrict.
// ---------------------------------------------------------------------------
__global__ void __launch_bounds__(512)
k_prod_linear(const float* __restrict__ Bs, const float* __restrict__ Bv,
              const float* __restrict__ Wps, const float* __restrict__ Wpv,
              const float* sc_s, const float* __restrict__ sc_v,
              float* __restrict__ dout, float* souts)
{
    const int tid  = threadIdx.x;
    const int lane = tid & 31;
    const int w    = tid >> 5;
    const int mat  = w >> 2;
    const int nt   = w & 3;
    const int m0   = blockIdx.x * 16;
    const int l15  = lane & 15;
    const int rb   = (lane < 16) ? 0 : 8;

    if (mat == 0) {
        v8f acc = wmma_tile_k64<true>(Bs + (size_t)m0 * FF, FF, 1,
                                      Wps + nt * 16, FF, lane);
#pragma unroll
        for (int r = 0; r < 8; ++r) {
            int row = m0 + rb + r, col = nt * 16 + l15;
            float val = acc[r] + sc_s[(size_t)row * FF + col];
            dout[NN + (size_t)row * FF + col] = val;
            souts[(size_t)row * FF + col]     = val;
        }
    } else {
        int c = mat - 1;
        v8f acc = wmma_tile_k64<true>(Bv + (size_t)c * NF + (size_t)m0 * FF, FF, 1,
                                      Wpv + nt * 16, FF, lane);
#pragma unroll
        for (int r = 0; r < 8; ++r) {
            int row = m0 + rb + r, col = nt * 16 + l15;
            float val = acc[r] + sc_v[(size_t)c * NF + (size_t)row * FF + col];
            dout[NN + (size_t)NF + ((size_t)row * FF + col) * 3 + c] = val;
        }
    }
}

// K4c: node_outputs = s_out @ W_read  -> d_out[0..N)
__global__ void __launch_bounds__(256)
k_readout(const float* __restrict__ souts, const float* __restrict__ Wr,
          float* __restrict__ dout)
{
    int n = blockIdx.x * 256 + threadIdx.x;
    if (n >= NN) return;
    float acc = 0.f;
    for (int g = 0; g < 64; ++g) acc += souts[(size_t)n * 64 + g] * Wr[g];
    dout[n] = acc;
}

// ---------------------------------------------------------------------------
extern "C" void kernel_launch(void* const* d_in, const int* in_sizes, int n_in,
                              void* d_out, int out_size, void* d_ws, size_t ws_size,
                              hipStream_t stream)
{
    (void)in_sizes; (void)n_in; (void)out_size; (void)ws_size;

    const float* s    = (const float*)d_in[0];
    const float* v    = (const float*)d_in[1];
    const float* Y1   = (const float*)d_in[2];
    const float* ef   = (const float*)d_in[3];
    const int*   spec = (const int*)d_in[4];
    const int*   snd  = (const int*)d_in[5];
    const int*   rcv  = (const int*)d_in[6];
    const float* Wls  = (const float*)d_in[7];
    const float* Wlv  = (const float*)d_in[8];
    const float* Wscs = (const float*)d_in[9];
    const float* Wscv = (const float*)d_in[10];
    const float* W1   = (const float*)d_in[11];
    const float* W2   = (const float*)d_in[12];
    const float* Wis  = (const float*)d_in[13];
    const float* Wiv  = (const float*)d_in[14];
    const float* wps  = (const float*)d_in[15];
    const float* wpv  = (const float*)d_in[16];
    const float* Wps  = (const float*)d_in[17];
    const float* Wpv  = (const float*)d_in[18];
    const float* Wr   = (const float*)d_in[19];
    float* out = (float*)d_out;

    float* ws   = (float*)d_ws;
    float* s1   = ws;                      // [NF]   -> reused for Asp
    float* v1   = ws + (size_t)NF;         // [3NF]  -> reused for Avp
    float* scs  = ws + (size_t)4 * NF;     // [NF]   -> reused for s_out copy
    float* scv  = ws + (size_t)5 * NF;     // [3NF]
    float* As   = ws + (size_t)8 * NF;     // [NF]   -> reused for B_s
    float* Av   = ws + (size_t)9 * NF;     // [3NF]  -> reused for B_v

    hipMemsetAsync(As, 0, (size_t)4 * NF * sizeof(float), stream);

    k_node_linear<<<NN / 16, 512, 0, stream>>>(s, v, Wls, Wlv, s1, v1);
    k_skip<<<(NF + 255) / 256, 256, 0, stream>>>(s, v, spec, Wscs, Wscv, scs, scv);
    k_edge<<<EE / EPB, 128, 0, stream>>>(Y1, ef, snd, rcv, W1, W2, s1, v1, As, Av);
    k_int_linear<<<NN / 16, 512, 0, stream>>>(As, Av, Wis, Wiv, s1, v1);
    k_product<<<(NF + 255) / 256, 256, 0, stream>>>(s1, v1, spec, wps, wpv, As, Av);
    k_prod_linear<<<NN / 16, 512, 0, stream>>>(As, Av, Wps, Wpv, scs, scv, out, scs);
    k_readout<<<(NN + 255) / 256, 256, 0, stream>>>(scs, Wr, out);
}